// GATBackbone_28492813041762
// MI455X (gfx1250) — compile-verified
//
#include <hip/hip_runtime.h>
#include <cstdint>
#include <cstddef>

// ---------------------------------------------------------------------------
// GAT backbone for MI455X (gfx1250).
//   feature dim F = H*C = 4*32 = 128 for every layer (input is also 128).
//   Dense h = X @ W done with V_WMMA_F32_16X16X4_F32 (fp32 WMMA, full precision).
//   Edge softmax/scatter phase is L2-resident (whole working set < 192 MB L2).
// ---------------------------------------------------------------------------

typedef __attribute__((ext_vector_type(2))) float v2f;
typedef __attribute__((ext_vector_type(8))) float v8f;

#define HEADS 4
#define CPH   32
#define FDIM  128
#define NGRAPHS 64

static inline int cdiv(long long a, long long b) { return (int)((a + b - 1) / b); }

__device__ __forceinline__ float lrelu(float x, float s) { return x > 0.f ? x : x * s; }

// order-preserving float <-> u32 map so segment-max can use native u32 atomic max
__device__ __forceinline__ unsigned int encOrd(float f) {
  unsigned int u = __float_as_uint(f);
  return (u & 0x80000000u) ? ~u : (u | 0x80000000u);
}
__device__ __forceinline__ float decOrd(unsigned int u) {
  return (u & 0x80000000u) ? __uint_as_float(u ^ 0x80000000u) : __uint_as_float(~u);
}
#define ENC_NEG_INF 0x007FFFFFu   // encOrd(-inf)

__device__ __forceinline__ void edge_pair(const int* __restrict__ ei, int e, int E,
                                          int& s, int& d) {
  if (e < E) { s = ei[e]; d = ei[E + e]; }
  else       { s = e - E; d = e - E; }          // self-loops appended after E edges
}

// --------------------------------------------------------------------------
// GEMM: Hout[M,128] = X[M,128] @ W[128,128], fp32 WMMA 16x16x4.
// block = 256 threads (8 waves). Each block: 16-row stripe of X (staged in LDS,
// contiguous 8KB), each wave one 16x16 tile -> 8 waves cover all 128 columns.
// --------------------------------------------------------------------------
__global__ __launch_bounds__(256) void gat_gemm_wmma(const float* __restrict__ X,
                                                     const float* __restrict__ W,
                                                     float* __restrict__ Hout,
                                                     int M) {
  __shared__ float xt[16 * FDIM];             // 8 KB
  const int t  = threadIdx.x;
  const int m0 = blockIdx.x * 16;

  // stage 16 consecutive rows = contiguous 2048 floats of X
  const float* xsrc = X + (size_t)m0 * FDIM;
#pragma unroll
  for (int i = 0; i < 8; ++i) {
    int idx = t + i * 256;
    int row = m0 + (idx >> 7);
    xt[idx] = (row < M) ? xsrc[idx] : 0.f;
  }
  __syncthreads();

  const int wave = t >> 5;                    // 0..7 -> output column tile
  const int lane = t & 31;
  const int l16  = lane & 15;
  const int hi   = lane >> 4;                 // 0 or 1
  const int col  = wave * 16 + l16;

  v8f acc = {};
#pragma unroll
  for (int k0 = 0; k0 < FDIM; k0 += 4) {
    const int ka = k0 + 2 * hi;               // A: VGPR0 holds K=ka, VGPR1 K=ka+1
    v2f a, b;
    a.x = xt[l16 * FDIM + ka];
    a.y = xt[l16 * FDIM + ka + 1];
    b.x = W[(size_t)ka * FDIM + col];         // B: row ka striped across lanes
    b.y = W[(size_t)(ka + 1) * FDIM + col];
    // 8 args: (neg_a, A, neg_b, B, c_mod, C, reuse_a, reuse_b)
    acc = __builtin_amdgcn_wmma_f32_16x16x4_f32(false, a, false, b,
                                                (short)0, acc, false, false);
  }

  // C/D layout: VGPR r -> row (r + 8*hi), col = lane%16
#pragma unroll
  for (int r = 0; r < 8; ++r) {
    int row = m0 + hi * 8 + r;
    if (row < M) Hout[(size_t)row * FDIM + col] = acc[r];
  }
}

// per-node attention scalars: s[n,h] = <h[n,h,:], a_src[h,:]>, same for d
__global__ void gat_attn_sd(const float* __restrict__ Hl,
                            const float* __restrict__ aS,
                            const float* __restrict__ aD,
                            float* __restrict__ sArr, float* __restrict__ dArr, int N) {
  int tid = blockIdx.x * blockDim.x + threadIdx.x;
  if (tid >= N * HEADS) return;
  int n = tid >> 2, hd = tid & 3;
  const float* hp = Hl + (size_t)n * FDIM + hd * CPH;
  const float* as = aS + hd * CPH;
  const float* ad = aD + hd * CPH;
  float ss = 0.f, dd = 0.f;
#pragma unroll
  for (int c = 0; c < CPH; ++c) { float v = hp[c]; ss += v * as[c]; dd += v * ad[c]; }
  sArr[tid] = ss;
  dArr[tid] = dd;
}

__global__ void gat_fill_u32(unsigned int* __restrict__ p, unsigned int v, int n) {
  int tid = blockIdx.x * blockDim.x + threadIdx.x;
  if (tid < n) p[tid] = v;
}

// pass 1: e = leaky_relu(s[src]+d[dst], 0.2); segment max over dst (encoded u32)
__global__ void gat_edge_max(const float* __restrict__ sArr, const float* __restrict__ dArr,
                             const int* __restrict__ ei, int E, int Etot,
                             float* __restrict__ ebuf, unsigned int* __restrict__ menc) {
  int tid = blockIdx.x * blockDim.x + threadIdx.x;
  if (tid >= Etot * HEADS) return;
  int e = tid >> 2, hd = tid & 3;
  int s, d; edge_pair(ei, e, E, s, d);
  float ev = lrelu(sArr[s * HEADS + hd] + dArr[d * HEADS + hd], 0.2f);
  ebuf[tid] = ev;
  atomicMax(&menc[d * HEADS + hd], encOrd(ev));
}

// pass 2: ex = exp(e - m[dst]); denom[dst] += ex
__global__ void gat_edge_exp(const int* __restrict__ ei, int E, int Etot,
                             float* __restrict__ ebuf,
                             const unsigned int* __restrict__ menc,
                             float* __restrict__ denom) {
  int tid = blockIdx.x * blockDim.x + threadIdx.x;
  if (tid >= Etot * HEADS) return;
  int e = tid >> 2, hd = tid & 3;
  int s, d; edge_pair(ei, e, E, s, d);
  float m  = decOrd(menc[d * HEADS + hd]);
  float ex = __expf(ebuf[tid] - m);
  ebuf[tid] = ex;
  atomicAdd(&denom[d * HEADS + hd], ex);
}

// pass 3: out[dst,f] += (ex/denom[dst]) * h[src,f]   (one thread per edge-feature)
__global__ void gat_edge_scatter(const int* __restrict__ ei, int E, int Etot,
                                 const float* __restrict__ ebuf,
                                 const float* __restrict__ denom,
                                 const float* __restrict__ Hl,
                                 float* __restrict__ out) {
  long long tid = (long long)blockIdx.x * blockDim.x + threadIdx.x;
  if (tid >= (long long)Etot * FDIM) return;
  int e  = (int)(tid >> 7);
  int f  = (int)(tid & (FDIM - 1));
  int hd = f >> 5;
  int s, d; edge_pair(ei, e, E, s, d);
  float alpha = ebuf[e * HEADS + hd] / denom[d * HEADS + hd];
  atomicAdd(&out[(size_t)d * FDIM + f], alpha * Hl[(size_t)s * FDIM + f]);
}

// out = leaky_relu(out + bias, 0.01)
__global__ void gat_bias_act(float* __restrict__ out, const float* __restrict__ b,
                             int total) {
  int tid = blockIdx.x * blockDim.x + threadIdx.x;
  if (tid >= total) return;
  float v = out[tid] + b[tid & (FDIM - 1)];
  out[tid] = lrelu(v, 0.01f);
}

// graph pooling: encoded atomic max per (graph, feature)
__global__ void gat_pool_max(const float* __restrict__ hfin, const int* __restrict__ batch,
                             unsigned int* __restrict__ embEnc, int N) {
  int tid = blockIdx.x * blockDim.x + threadIdx.x;
  if (tid >= N * FDIM) return;
  int n = tid >> 7, f = tid & (FDIM - 1);
  int g = batch[n];
  atomicMax(&embEnc[g * FDIM + f], encOrd(hfin[tid]));
}

__global__ void gat_decode_emb(const unsigned int* __restrict__ enc,
                               float* __restrict__ out, int n) {
  int tid = blockIdx.x * blockDim.x + threadIdx.x;
  if (tid < n) out[tid] = decOrd(enc[tid]);
}

__global__ void gat_copy_ei(const int* __restrict__ ei, float* __restrict__ out, int n) {
  int tid = blockIdx.x * blockDim.x + threadIdx.x;
  if (tid < n) out[tid] = (float)ei[tid];
}

// ---------------------------------------------------------------------------

extern "C" void kernel_launch(void* const* d_in, const int* in_sizes, int n_in,
                              void* d_out, int out_size, void* d_ws, size_t ws_size,
                              hipStream_t stream) {
  const float* x     = (const float*)d_in[0];
  const int*   ei    = (const int*)d_in[1];
  const int*   batch = (const int*)d_in[2];
  const float* Wl[3] = {(const float*)d_in[3], (const float*)d_in[7],  (const float*)d_in[11]};
  const float* aS[3] = {(const float*)d_in[4], (const float*)d_in[8],  (const float*)d_in[12]};
  const float* aD[3] = {(const float*)d_in[5], (const float*)d_in[9],  (const float*)d_in[13]};
  const float* bL[3] = {(const float*)d_in[6], (const float*)d_in[10], (const float*)d_in[14]};

  const int N    = in_sizes[0] / FDIM;     // 50000
  const int E    = in_sizes[1] / 2;        // 800000
  const int Etot = E + N;                  // edges incl. self-loops

  // workspace layout (~68 MB): bufH | feat | s | d | menc | denom | ebuf | embEnc
  float*        bufH   = (float*)d_ws;                          // N*F  (linear output)
  float*        feat   = bufH + (size_t)N * FDIM;               // N*F  (layer in/out ping)
  float*        sArr   = feat + (size_t)N * FDIM;               // N*H
  float*        dArr   = sArr + (size_t)N * HEADS;              // N*H
  unsigned int* menc   = (unsigned int*)(dArr + (size_t)N * HEADS);   // N*H
  float*        denom  = (float*)(menc + (size_t)N * HEADS);    // N*H
  float*        ebuf   = denom + (size_t)N * HEADS;             // Etot*H
  unsigned int* embEnc = (unsigned int*)(ebuf + (size_t)Etot * HEADS); // G*F

  // output layout: [ graph_embedding G*F | h N*F | edge_index 2E ]
  float* emb   = (float*)d_out;
  float* hOut  = emb + NGRAPHS * FDIM;
  float* eiOut = hOut + (size_t)N * FDIM;

  const float* layerIn[3]  = {x,    feat, feat};
  float*       layerOut[3] = {feat, feat, hOut};

  const int NH = N * HEADS;

  for (int L = 0; L < 3; ++L) {
    // 1) h = X @ W  (fp32 WMMA)
    gat_gemm_wmma<<<cdiv(N, 16), 256, 0, stream>>>(layerIn[L], Wl[L], bufH, N);
    // 2) attention scalars
    gat_attn_sd<<<cdiv(NH, 256), 256, 0, stream>>>(bufH, aS[L], aD[L], sArr, dArr, N);
    // 3) init softmax state + zero the aggregation buffer
    gat_fill_u32<<<cdiv(NH, 256), 256, 0, stream>>>(menc, ENC_NEG_INF, NH);
    hipMemsetAsync(denom, 0, (size_t)NH * sizeof(float), stream);
    hipMemsetAsync(layerOut[L], 0, (size_t)N * FDIM * sizeof(float), stream);
    // 4) segment softmax + scatter
    gat_edge_max<<<cdiv((long long)Etot * HEADS, 256), 256, 0, stream>>>(
        sArr, dArr, ei, E, Etot, ebuf, menc);
    gat_edge_exp<<<cdiv((long long)Etot * HEADS, 256), 256, 0, stream>>>(
        ei, E, Etot, ebuf, menc, denom);
    gat_edge_scatter<<<cdiv((long long)Etot * FDIM, 256), 256, 0, stream>>>(
        ei, E, Etot, ebuf, denom, bufH, layerOut[L]);
    // 5) bias + LeakyReLU(0.01)
    gat_bias_act<<<cdiv((long long)N * FDIM, 256), 256, 0, stream>>>(
        layerOut[L], bL[L], N * FDIM);
  }

  // graph max-pool over batch
  gat_fill_u32<<<cdiv(NGRAPHS * FDIM, 256), 256, 0, stream>>>(embEnc, ENC_NEG_INF,
                                                              NGRAPHS * FDIM);
  gat_pool_max<<<cdiv((long long)N * FDIM, 256), 256, 0, stream>>>(hOut, batch, embEnc, N);
  gat_decode_emb<<<cdiv(NGRAPHS * FDIM, 256), 256, 0, stream>>>(embEnc, emb,
                                                                NGRAPHS * FDIM);
  // pass edge_index through (values < 2^24, exactly representable as float)
  gat_copy_ei<<<cdiv(2LL * E, 256), 256, 0, stream>>>(ei, eiOut, 2 * E);
}